// GAT_47897475285259
// MI455X (gfx1250) — compile-verified
//
#include <hip/hip_runtime.h>
#include <hip/hip_bf16.h>
#include <math.h>

// ---------------- problem constants (match reference) ----------------
#define NUM_USER 50000
#define NUM_ITEM 50000
#define N_NODES  (NUM_USER + NUM_ITEM)   // 100000
#define DIM      64
#define HEADS    3
#define HDIM     (HEADS * DIM)           // 192
#define E_BASE   500000
#define E_TOT    (2 * E_BASE + N_NODES)  // 1,100,000 (fwd + rev + self loops)
#define BATCH    4096
#define MTILE    5                       // row tiles per wave: 100000/(16*5)=1250 exact

// ---------------- workspace layout (bytes) ----------------
#define X_OFF    0ull                                          // x: [N, 192] f32
#define ASRC_OFF (X_OFF    + (size_t)N_NODES * HDIM * 4)       // a_src: [N,3]
#define ADST_OFF (ASRC_OFF + (size_t)N_NODES * HEADS * 4)      // a_dst: [N,3]
#define SMAX_OFF (ADST_OFF + (size_t)N_NODES * HEADS * 4)      // seg max (encoded u32): [N,3]
#define DEN_OFF  (SMAX_OFF + (size_t)N_NODES * HEADS * 4)      // softmax denom: [N,3]
#define EVAL_OFF (DEN_OFF  + (size_t)N_NODES * HEADS * 4)      // per-edge exp: [E,3]
#define ACC_OFF  (EVAL_OFF + (size_t)E_TOT   * HEADS * 4)      // head-mean accum: [N,64]

typedef float v2f __attribute__((ext_vector_type(2)));
typedef float v8f __attribute__((ext_vector_type(8)));

// ---------------- helpers ----------------
// order-preserving f32 <-> u32 encoding so atomicMax(u32) == float max
__device__ __forceinline__ unsigned int fenc(float f) {
  unsigned int u = __float_as_uint(f);
  return (u & 0x80000000u) ? ~u : (u | 0x80000000u);
}
__device__ __forceinline__ float fdec(unsigned int e) {
  return (e & 0x80000000u) ? __uint_as_float(e & 0x7FFFFFFFu)
                           : __uint_as_float(~e);
}

// decode edge e of the full (fwd | reversed | self-loop) edge set
__device__ __forceinline__ void edge_sd(const int* __restrict__ ei, int e,
                                        int& s, int& d) {
  if (e < E_BASE) {              // original user->item edges
    s = ei[e];
    d = ei[E_BASE + e];
  } else if (e < 2 * E_BASE) {   // reversed item->user edges
    s = ei[E_BASE + (e - E_BASE)];
    d = ei[e - E_BASE];
  } else {                       // self loops
    s = d = e - 2 * E_BASE;
  }
}

// ---------------- kernel 1: x = emb @ W via v_wmma_f32_16x16x4_f32 ----------------
// one wave computes MTILE 16x16 output tiles stacked in M, reusing each
// B fragment across MTILE WMMAs; K=64 in 16 steps of 4
__global__ __launch_bounds__(32)
void gat_gemm_wmma(const float* __restrict__ emb, const float* __restrict__ W,
                   float* __restrict__ x) {
  const int lane = threadIdx.x;            // 0..31
  const int half = lane >> 4;              // lane groups 0-15 / 16-31
  const int m    = lane & 15;
  const int row0 = blockIdx.x * (16 * MTILE);
  const int col0 = blockIdx.y * 16;        // over HDIM (192/16 = 12 exact)

  const float* __restrict__ bcol = W + col0 + m;

  v8f c[MTILE];
#pragma unroll
  for (int t = 0; t < MTILE; ++t) c[t] = (v8f){0.f,0.f,0.f,0.f,0.f,0.f,0.f,0.f};

#pragma unroll 4
  for (int k = 0; k < DIM; k += 4) {
    const int ka = k + 2 * half;           // ISA A layout: lanes16-31 hold K+2,K+3
    v2f b;
    b.x = bcol[(size_t)ka * HDIM];         // B 4x16: rows K striped like A halves
    b.y = bcol[(size_t)(ka + 1) * HDIM];
#pragma unroll
    for (int t = 0; t < MTILE; ++t) {
      const float* __restrict__ arow =
          emb + (size_t)(row0 + t * 16 + m) * DIM;
      v2f a;
      a.x = arow[ka];
      a.y = arow[ka + 1];
      c[t] = __builtin_amdgcn_wmma_f32_16x16x4_f32(
          /*neg_a=*/false, a, /*neg_b=*/false, b,
          /*c_mod=*/(short)0, c[t], /*reuse_a=*/false, /*reuse_b=*/false);
    }
  }
  // C/D layout: VGPR i -> row M=i (lanes 0-15) / M=i+8 (lanes 16-31), col N=m
#pragma unroll
  for (int t = 0; t < MTILE; ++t)
#pragma unroll
    for (int i = 0; i < 8; ++i)
      x[(size_t)(row0 + t * 16 + i + 8 * half) * HDIM + col0 + m] = c[t][i];
}

// ---------------- kernel 2: a_src[n,h], a_dst[n,h] ----------------
__global__ void gat_attn_coef(const float* __restrict__ x,
                              const float* __restrict__ att_src,
                              const float* __restrict__ att_dst,
                              float* __restrict__ a_src,
                              float* __restrict__ a_dst) {
  const int tid = blockIdx.x * blockDim.x + threadIdx.x;
  if (tid >= N_NODES * HEADS) return;
  const int n = tid / HEADS, h = tid % HEADS;
  const float4* __restrict__ xv = (const float4*)(x + (size_t)n * HDIM + h * DIM);
  const float4* __restrict__ sv = (const float4*)(att_src + h * DIM);
  const float4* __restrict__ dv = (const float4*)(att_dst + h * DIM);
  float s = 0.f, d = 0.f;
#pragma unroll
  for (int i = 0; i < DIM / 4; ++i) {
    const float4 xx = xv[i], aa = sv[i], bb = dv[i];
    s += xx.x * aa.x + xx.y * aa.y + xx.z * aa.z + xx.w * aa.w;
    d += xx.x * bb.x + xx.y * bb.y + xx.z * bb.z + xx.w * bb.w;
  }
  a_src[tid] = s;
  a_dst[tid] = d;
}

// ---------------- kernel 3: segment max of leaky-relu logits ----------------
__global__ void gat_edge_max(const int* __restrict__ ei,
                             const float* __restrict__ a_src,
                             const float* __restrict__ a_dst,
                             unsigned int* __restrict__ smax) {
  const int e = blockIdx.x * blockDim.x + threadIdx.x;
  if (e >= E_TOT) return;
  int s, d;
  edge_sd(ei, e, s, d);
#pragma unroll
  for (int h = 0; h < HEADS; ++h) {
    float l = a_src[s * HEADS + h] + a_dst[d * HEADS + h];
    l = (l > 0.f) ? l : 0.2f * l;           // leaky_relu slope 0.2
    atomicMax(&smax[d * HEADS + h], fenc(l));
  }
}

// ---------------- kernel 4: e = exp(l - max); denom += e ----------------
__global__ void gat_edge_exp(const int* __restrict__ ei,
                             const float* __restrict__ a_src,
                             const float* __restrict__ a_dst,
                             const unsigned int* __restrict__ smax,
                             float* __restrict__ den,
                             float* __restrict__ evals) {
  const int e = blockIdx.x * blockDim.x + threadIdx.x;
  if (e >= E_TOT) return;
  int s, d;
  edge_sd(ei, e, s, d);
#pragma unroll
  for (int h = 0; h < HEADS; ++h) {
    float l = a_src[s * HEADS + h] + a_dst[d * HEADS + h];
    l = (l > 0.f) ? l : 0.2f * l;
    const float mx = fdec(smax[d * HEADS + h]); // finite: self-loops guarantee >=1 edge
    const float ev = expf(l - mx);
    evals[(size_t)e * HEADS + h] = ev;
    atomicAdd(&den[d * HEADS + h], ev);
  }
}

// ---------------- kernel 5: acc[d,f] += (1/H) * sum_h alpha * x[s,h,f] ----------------
// one thread per (edge, feature-quad): float4 gathers, 4 atomics; per-edge
// scalars (indices, evals, denoms) loaded 16x instead of 64x per edge
__global__ void gat_edge_agg(const int* __restrict__ ei,
                             const float* __restrict__ x,
                             const float* __restrict__ evals,
                             const float* __restrict__ den,
                             float* __restrict__ acc) {
  const int tid = blockIdx.x * blockDim.x + threadIdx.x;  // E_TOT*16 threads
  const int e = tid >> 4;
  const int f = (tid & 15) << 2;                          // 0,4,...,60
  if (e >= E_TOT) return;
  int s, d;
  edge_sd(ei, e, s, d);
  const float* __restrict__ xrow = x + (size_t)s * HDIM + f;
  float4 v = {0.f, 0.f, 0.f, 0.f};
#pragma unroll
  for (int h = 0; h < HEADS; ++h) {
    const float alpha =
        evals[(size_t)e * HEADS + h] / (den[d * HEADS + h] + 1e-16f);
    const float4 xv = *(const float4*)(xrow + h * DIM);
    v.x += alpha * xv.x;
    v.y += alpha * xv.y;
    v.z += alpha * xv.z;
    v.w += alpha * xv.w;
  }
  float* __restrict__ a = acc + (size_t)d * DIM + f;
  const float sc = 1.0f / HEADS;
  atomicAdd(a + 0, v.x * sc);
  atomicAdd(a + 1, v.y * sc);
  atomicAdd(a + 2, v.z * sc);
  atomicAdd(a + 3, v.w * sc);
}

// ---------------- kernel 6: in-place bias + leaky_relu(0.01), float4 ----------------
__global__ void gat_finalize(float* __restrict__ acc,
                             const float* __restrict__ bias) {
  const int tid = blockIdx.x * blockDim.x + threadIdx.x;  // N*16 threads (exact)
  const float4 b = ((const float4*)bias)[tid & 15];
  float4 v = ((float4*)acc)[tid];
  v.x += b.x; v.y += b.y; v.z += b.z; v.w += b.w;
  v.x = (v.x > 0.f) ? v.x : 0.01f * v.x;
  v.y = (v.y > 0.f) ? v.y : 0.01f * v.y;
  v.z = (v.z > 0.f) ? v.z : 0.01f * v.z;
  v.w = (v.w > 0.f) ? v.w : 0.01f * v.w;
  ((float4*)acc)[tid] = v;
}

// ---------------- kernel 7: BPR scores ----------------
__global__ void gat_score(const int* __restrict__ un, const int* __restrict__ pn,
                          const int* __restrict__ nn,
                          const float* __restrict__ xf, float* __restrict__ out) {
  const int b = blockIdx.x * blockDim.x + threadIdx.x;
  if (b >= BATCH) return;
  const float4* __restrict__ u = (const float4*)(xf + (size_t)un[b] * DIM);
  const float4* __restrict__ p = (const float4*)(xf + (size_t)pn[b] * DIM);
  const float4* __restrict__ q = (const float4*)(xf + (size_t)nn[b] * DIM);
  float ps = 0.f, ns = 0.f;
#pragma unroll
  for (int i = 0; i < DIM / 4; ++i) {
    const float4 uv = u[i], pv = p[i], qv = q[i];
    ps += uv.x * pv.x + uv.y * pv.y + uv.z * pv.z + uv.w * pv.w;
    ns += uv.x * qv.x + uv.y * qv.y + uv.z * qv.z + uv.w * qv.w;
  }
  out[b] = ps;          // pos_scores
  out[BATCH + b] = ns;  // neg_scores
}

// ---------------- launcher ----------------
extern "C" void kernel_launch(void* const* d_in, const int* in_sizes, int n_in,
                              void* d_out, int out_size, void* d_ws,
                              size_t ws_size, hipStream_t stream) {
  const int*   user_nodes = (const int*)d_in[0];
  const int*   pos_nodes  = (const int*)d_in[1];
  const int*   neg_nodes  = (const int*)d_in[2];
  const float* emb        = (const float*)d_in[3];   // [N, 64]
  const int*   ei         = (const int*)d_in[4];     // [2, E_BASE]
  const float* W          = (const float*)d_in[5];   // [64, 192]
  const float* att_src    = (const float*)d_in[6];   // [3, 64]
  const float* att_dst    = (const float*)d_in[7];   // [3, 64]
  const float* bias       = (const float*)d_in[8];   // [64]
  float* out = (float*)d_out;

  char* ws = (char*)d_ws;
  float*        x    = (float*)(ws + X_OFF);
  float*        asrc = (float*)(ws + ASRC_OFF);
  float*        adst = (float*)(ws + ADST_OFF);
  unsigned int* smax = (unsigned int*)(ws + SMAX_OFF);
  float*        den  = (float*)(ws + DEN_OFF);
  float*        ev   = (float*)(ws + EVAL_OFF);
  float*        acc  = (float*)(ws + ACC_OFF);

  // zero the reduction buffers (memset nodes are graph-capture safe)
  hipMemsetAsync(smax, 0, (size_t)N_NODES * HEADS * 4, stream); // enc-sentinel < any real
  hipMemsetAsync(den,  0, (size_t)N_NODES * HEADS * 4, stream);
  hipMemsetAsync(acc,  0, (size_t)N_NODES * DIM * 4, stream);

  dim3 ggrid(N_NODES / (16 * MTILE), HDIM / 16);
  gat_gemm_wmma<<<ggrid, 32, 0, stream>>>(emb, W, x);

  gat_attn_coef<<<(N_NODES * HEADS + 255) / 256, 256, 0, stream>>>(
      x, att_src, att_dst, asrc, adst);

  gat_edge_max<<<(E_TOT + 255) / 256, 256, 0, stream>>>(ei, asrc, adst, smax);
  gat_edge_exp<<<(E_TOT + 255) / 256, 256, 0, stream>>>(ei, asrc, adst, smax,
                                                        den, ev);
  gat_edge_agg<<<(E_TOT * 16) / 256, 256, 0, stream>>>(ei, x, ev, den, acc);

  gat_finalize<<<(N_NODES * 16) / 256, 256, 0, stream>>>(acc, bias);
  gat_score<<<(BATCH + 255) / 256, 256, 0, stream>>>(user_nodes, pos_nodes,
                                                     neg_nodes, acc, out);
}